// TensorDLT_83700322664812
// MI455X (gfx1250) — compile-verified
//
#include <hip/hip_runtime.h>

// ---------------------------------------------------------------------------
// TensorDLT: batch-131072 4-point homography solve h = A^-1 b.
// Columns 0-5 of A are batch-invariant (built only from corners_a), so the
// 8x8 solve reduces analytically to a per-lane 2x2 solve for (h6,h7) plus
// two uniform-adjugate 3x3 back-substitutions -> ~100 VALU ops/lane.
// Memory-bound: ~8 MB traffic @ 23.3 TB/s ~= 0.36 us.
// CDNA5 paths: wave32 blocks, global_load_async_to_lds_b128 (ASYNCcnt),
// s_wait_asynccnt, ds_load_b128 readback, nontemporal b128 stores.
// WMMA is inapplicable: every lane owns a distinct system; CDNA5 WMMA is one
// wave-striped 16x16 product per wave (no shared-operand GEMM exists here).
// ---------------------------------------------------------------------------

#ifndef __has_builtin
#define __has_builtin(x) 0
#endif

#define AS1 __attribute__((address_space(1)))
#define AS3 __attribute__((address_space(3)))

#if __has_builtin(__builtin_amdgcn_global_load_async_to_lds_b128)
#define USE_ASYNC_BUILTIN 1
#else
#define USE_ASYNC_BUILTIN 0
#endif

typedef float v4f __attribute__((ext_vector_type(4)));
typedef int   v4i __attribute__((ext_vector_type(4)));

#define TPB 256

__global__ __launch_bounds__(TPB) void tensor_dlt_kernel(
    const float* __restrict__ corners_a,
    const float* __restrict__ preds,
    float* __restrict__ out,
    int n)
{
    __shared__ __align__(16) float tile[TPB * 8];   // 8 KB: this block's preds rows

    const int tid = threadIdx.x;
    const int i   = blockIdx.x * TPB + tid;

    // ---- async stage: 32 B/lane of preds -> LDS (dedicated async path) ----
    if (i < n) {
        const float* gsrc = preds + (size_t)i * 8;
#if USE_ASYNC_BUILTIN
        __builtin_amdgcn_global_load_async_to_lds_b128(
            (AS1 v4i*)gsrc, (AS3 v4i*)&tile[tid * 8], 0, 0);
        __builtin_amdgcn_global_load_async_to_lds_b128(
            (AS1 v4i*)gsrc, (AS3 v4i*)&tile[tid * 8], 16, 0);
#else
        unsigned lds = (unsigned)(unsigned long long)(AS3 void*)&tile[tid * 8];
        unsigned long long ga = (unsigned long long)gsrc;
        asm volatile(
            "global_load_async_to_lds_b128 %0, %1, off\n\t"
            "global_load_async_to_lds_b128 %0, %1, off offset:16"
            :: "v"(lds), "v"(ga) : "memory");
#endif
    }
#if __has_builtin(__builtin_amdgcn_s_wait_asynccnt)
    __builtin_amdgcn_s_wait_asynccnt(0);
#else
    asm volatile("s_wait_asynccnt 0x0" ::: "memory");
#endif
    // Each lane consumes only the bytes it staged itself -> ASYNCcnt==0
    // suffices, no workgroup barrier needed.

    if (i >= n) return;

    // ---- uniform source corners (lowered to SMEM s_load_b256) ----
    const v4f c01 = ((const v4f*)corners_a)[0];
    const v4f c23 = ((const v4f*)corners_a)[1];
    const float ua[4] = {c01.x, c01.z, c23.x, c23.z};
    const float va[4] = {c01.y, c01.w, c23.y, c23.w};

    // ---- uniform precompute (wave-invariant, ~40 ops) ----
    // Null vector w of the 4x3 matrix C = [ua va 1]: w_j = (-1)^j det(minor_j)
    const float D0 = (ua[2]-ua[1])*(va[3]-va[1]) - (ua[3]-ua[1])*(va[2]-va[1]);
    const float D1 = (ua[2]-ua[0])*(va[3]-va[0]) - (ua[3]-ua[0])*(va[2]-va[0]);
    const float D2 = (ua[1]-ua[0])*(va[3]-va[0]) - (ua[3]-ua[0])*(va[1]-va[0]);
    const float D3 = (ua[1]-ua[0])*(va[2]-va[0]) - (ua[2]-ua[0])*(va[1]-va[0]);
    const float w[4]  = { D0, -D1, D2, -D3 };
    const float wu[4] = { w[0]*ua[0], w[1]*ua[1], w[2]*ua[2], w[3]*ua[3] };
    const float wv[4] = { w[0]*va[0], w[1]*va[1], w[2]*va[2], w[3]*va[3] };
    // Adjugate of the 3x3 [c0;c1;c2] (det = D3) for recovering h0..h2 / h3..h5
    const float A11 = va[1]-va[2], A12 = va[2]-va[0], A13 = va[0]-va[1];
    const float A21 = ua[2]-ua[1], A22 = ua[0]-ua[2], A23 = ua[1]-ua[0];
    const float A31 = ua[1]*va[2]-ua[2]*va[1];
    const float A32 = ua[2]*va[0]-ua[0]*va[2];
    const float A33 = ua[0]*va[1]-ua[1]*va[0];
    const float inv3 = 1.0f / D3;

    // ---- per-lane: pull preds row back from LDS (ds_load_b128) ----
    const v4f p01 = *(const v4f*)&tile[tid * 8];
    const v4f p23 = *(const v4f*)&tile[tid * 8 + 4];
    const float pb[8] = {p01.x, p01.y, p01.z, p01.w,
                         p23.x, p23.y, p23.z, p23.w};

    float ub[4], vb[4];
#pragma unroll
    for (int j = 0; j < 4; ++j) {
        ub[j] = ua[j] + pb[2*j];
        vb[j] = va[j] + pb[2*j + 1];
    }

    // ---- 2x2 system for (h6, h7):
    //   (Σ wu·ub) h6 + (Σ wv·ub) h7 =  Σ w·ub
    //   (Σ wu·vb) h6 + (Σ wv·vb) h7 = -Σ w·vb
    float a11 = 0.f, a12 = 0.f, r1 = 0.f;
    float a21 = 0.f, a22 = 0.f, r2 = 0.f;
#pragma unroll
    for (int j = 0; j < 4; ++j) {
        a11 = __builtin_fmaf(wu[j], ub[j], a11);
        a12 = __builtin_fmaf(wv[j], ub[j], a12);
        r1  = __builtin_fmaf(w[j],  ub[j], r1);
        a21 = __builtin_fmaf(wu[j], vb[j], a21);
        a22 = __builtin_fmaf(wv[j], vb[j], a22);
        r2  = __builtin_fmaf(-w[j], vb[j], r2);
    }
    const float idet2 = 1.0f / __builtin_fmaf(a11, a22, -(a12 * a21));
    const float h6 = (__builtin_fmaf(r1, a22, -(a12 * r2))) * idet2;
    const float h7 = (__builtin_fmaf(a11, r2, -(a21 * r1))) * idet2;

    // ---- recover (h0,h1,h2) and (h3,h4,h5) via uniform adjugate ----
    float t[3], s[3];
#pragma unroll
    for (int j = 0; j < 3; ++j) {
        const float g = __builtin_fmaf(ua[j], h6, va[j] * h7);
        t[j] = ub[j] * (g - 1.0f);
        s[j] = vb[j] * (g + 1.0f);
    }
    const float h0 = (__builtin_fmaf(A11, t[0], __builtin_fmaf(A12, t[1], A13 * t[2]))) * inv3;
    const float h1 = (__builtin_fmaf(A21, t[0], __builtin_fmaf(A22, t[1], A23 * t[2]))) * inv3;
    const float h2 = (__builtin_fmaf(A31, t[0], __builtin_fmaf(A32, t[1], A33 * t[2]))) * inv3;
    const float h3 = (__builtin_fmaf(A11, s[0], __builtin_fmaf(A12, s[1], A13 * s[2]))) * inv3;
    const float h4 = (__builtin_fmaf(A21, s[0], __builtin_fmaf(A22, s[1], A23 * s[2]))) * inv3;
    const float h5 = (__builtin_fmaf(A31, s[0], __builtin_fmaf(A32, s[1], A33 * s[2]))) * inv3;

    // ---- write-once output: nontemporal 128-bit stores ----
    v4f o0 = {h0, h1, h2, h3};
    v4f o1 = {h4, h5, h6, h7};
    __builtin_nontemporal_store(o0, (v4f*)(out + (size_t)i * 8));
    __builtin_nontemporal_store(o1, (v4f*)(out + (size_t)i * 8 + 4));
}

extern "C" void kernel_launch(void* const* d_in, const int* in_sizes, int n_in,
                              void* d_out, int out_size, void* d_ws, size_t ws_size,
                              hipStream_t stream) {
    const float* corners_a = (const float*)d_in[0];   // (8,)  f32
    const float* preds     = (const float*)d_in[1];   // (B,8) f32
    float*       out       = (float*)d_out;           // (B,8) f32

    const int n = in_sizes[1] / 8;
    const int blocks = (n + TPB - 1) / TPB;
    tensor_dlt_kernel<<<blocks, TPB, 0, stream>>>(corners_a, preds, out, n);
}